// PaiNNModule_21268678050218
// MI455X (gfx1250) — compile-verified
//
#include <hip/hip_runtime.h>
#include <math.h>

// Problem constants (match reference)
#define Nn 32768
#define En 524288
#define Hh 128
#define Rr 64

#define INV_SQRT_3f 0.57735026918962576f
#define INV_SQRT_Hf 0.088388347648318447f   // 1/sqrt(128)
#define INV_SQRT_2f 0.70710678118654752f

typedef float v2f __attribute__((ext_vector_type(2)));
typedef float v8f __attribute__((ext_vector_type(8)));

// V_WMMA_F32_16X16X4_F32: full-precision f32 matrix pipe (D = A(16x4) * B(4x16) + C)
__device__ __forceinline__ v8f wmma4(v2f a, v2f b, v8f c) {
  return __builtin_amdgcn_wmma_f32_16x16x4_f32(
      /*neg_a=*/false, a, /*neg_b=*/false, b,
      /*c_mod=*/(short)0, c, /*reuse_a=*/false, /*reuse_b=*/false);
}

// Hardware fp32 global atomic add (global_atomic_add_f32), avoids CAS loops.
__device__ __forceinline__ void atom_add(float* p, float v) { unsafeAtomicAdd(p, v); }

__device__ __forceinline__ float scaled_silu(float v) {
  return v * (1.0f / 0.6f) / (1.0f + __expf(-v));
}

// ---------------------------------------------------------------------------
// Generic f32 WMMA GEMM: C[M,NCOL] = A[M,K] * B[K,NCOL] (+bias) (optional silu)
// One wave computes a 16x64 output tile (4 accumulators); A fragment reused by
// 4 WMMAs. K/NCOL are compile-time: immediate address offsets, full unroll,
// and batched B-fragment loads so the scheduler overlaps loads with WMMAs.
// ---------------------------------------------------------------------------
template <int K, int NCOL, int SILU>
__global__ void gemm_wmma_f32(const float* __restrict__ A, const float* __restrict__ B,
                              const float* __restrict__ bias, float* __restrict__ C,
                              int M)
{
  const int lane = threadIdx.x & 31;
  const int wid  = threadIdx.x >> 5;
  constexpr int TN = NCOL / 64;                     // 64-col groups
  const long tile = (long)blockIdx.x * (blockDim.x >> 5) + wid;
  if (tile >= (long)(M >> 4) * TN) return;          // wave-uniform: EXEC stays all-ones
  const int mt = (int)(tile / TN);
  const int ng = (int)(tile % TN);
  const int half = lane >> 4;                       // 0 or 1
  const int l15  = lane & 15;
  const int row  = mt * 16 + l15;                   // A-matrix M index for this lane
  const int col0 = ng * 64 + l15;                   // first of 4 B/C columns for this lane
  const float* Arow = A + (long)row * K + half * 2; // A lane layout: K = 2*(lane>>4)+j
  const float* Bp   = B + (long)(half * 2) * NCOL + col0;

  v8f acc[4] = {};
#pragma unroll
  for (int k = 0; k < K; k += 4) {
    v2f a = *(const v2f*)(Arow + k);                // 8B-aligned
    v2f b[4];
#pragma unroll
    for (int t = 0; t < 4; ++t) {                   // batch all 4 B fragments first
      b[t].x = Bp[t * 16];
      b[t].y = Bp[NCOL + t * 16];
    }
#pragma unroll
    for (int t = 0; t < 4; ++t) acc[t] = wmma4(a, b[t], acc[t]);
    Bp += 4 * NCOL;
  }

  const int mbase = mt * 16 + half * 8;             // D layout: M = i + 8*(lane>>4)
#pragma unroll
  for (int t = 0; t < 4; ++t) {
    const int col = col0 + t * 16;
    const float bv = bias ? bias[col] : 0.0f;
#pragma unroll
    for (int i = 0; i < 8; ++i) {
      float v = acc[t][i] + bv;
      if (SILU) v = scaled_silu(v);
      C[(long)(mbase + i) * NCOL + col] = v;
    }
  }
}

// ---------------------------------------------------------------------------
// Fused edge kernel: rbfh tile(16 edges x 64 cols) = edge_embed @ Wr via WMMA,
// then epilogue m = xh[src]*rbfh and scatter-add with hw f32 atomics (L2-
// resident: 192MB L2 >> 50MB xh + 48MB vec + 67MB accumulators).
// 64-col groups never straddle a 128-col section boundary, so the scatter
// branch is wave-uniform and EXEC is all-ones around every WMMA.
// ---------------------------------------------------------------------------
__global__ void edge_msg_wmma(const float* __restrict__ eemb,
                              const float* __restrict__ Wr,
                              const float* __restrict__ br,
                              const int*   __restrict__ eidx,   // [2,E]
                              const float* __restrict__ evec,   // [E,3]
                              const float* __restrict__ xh,     // [N,384]
                              const float* __restrict__ vin,    // [N,3,128] (original vec)
                              float* __restrict__ xacc,         // [N,128]  (init = x)
                              float* __restrict__ vacc)         // [N,3,128](init = vec)
{
  const int lane = threadIdx.x & 31;
  const int wid  = threadIdx.x >> 5;
  const long tile = (long)blockIdx.x * (blockDim.x >> 5) + wid;
  if (tile >= (long)(En >> 4) * 6) return;          // 6 groups of 64 cols
  const int et = (int)(tile / 6);
  const int ng = (int)(tile % 6);
  const int half = lane >> 4;
  const int l15  = lane & 15;
  const int erow = et * 16 + l15;
  const int col0 = ng * 64 + l15;                   // 0..383
  const float* Arow = eemb + (long)erow * Rr + half * 2;
  const float* Bp   = Wr + (long)(half * 2) * 384 + col0;

  v8f acc[4] = {};
#pragma unroll
  for (int k = 0; k < Rr; k += 4) {
    v2f a = *(const v2f*)(Arow + k);
    v2f b[4];
#pragma unroll
    for (int t = 0; t < 4; ++t) {
      b[t].x = Bp[t * 16];
      b[t].y = Bp[384 + t * 16];
    }
#pragma unroll
    for (int t = 0; t < 4; ++t) acc[t] = wmma4(a, b[t], acc[t]);
    Bp += 4 * 384;
  }

  const int sec = ng >> 1;                          // 0: xh1, 1: xh2, 2: xh3 (wave-uniform)
  float bb[4];
#pragma unroll
  for (int t = 0; t < 4; ++t) bb[t] = br[col0 + t * 16];

  const int ebase = et * 16 + half * 8;
#pragma unroll
  for (int i = 0; i < 8; ++i) {
    const int e = ebase + i;
    const int s = eidx[e];
    const int d = eidx[En + e];
    const float* xhs = xh + (long)s * 384;
    if (sec == 0) {
#pragma unroll
      for (int t = 0; t < 4; ++t) {
        const int col = col0 + t * 16;
        const float m = xhs[col] * (acc[t][i] + bb[t]);
        atom_add(&xacc[(long)d * Hh + col], m);     // col == h in section 0
      }
    } else if (sec == 1) {
#pragma unroll
      for (int t = 0; t < 4; ++t) {
        const int col = col0 + t * 16;
        const int h = col & 127;
        const float m = xhs[col] * (acc[t][i] + bb[t]);
        const float tt = m * (INV_SQRT_3f * INV_SQRT_Hf);
#pragma unroll
        for (int c2 = 0; c2 < 3; ++c2)
          atom_add(&vacc[((long)d * 3 + c2) * Hh + h],
                   vin[((long)s * 3 + c2) * Hh + h] * tt);
      }
    } else {
      float ev[3];
#pragma unroll
      for (int c2 = 0; c2 < 3; ++c2) ev[c2] = evec[e * 3 + c2];
#pragma unroll
      for (int t = 0; t < 4; ++t) {
        const int col = col0 + t * 16;
        const int h = col & 127;
        const float m = xhs[col] * (acc[t][i] + bb[t]);
        const float tt = m * INV_SQRT_Hf;
#pragma unroll
        for (int c2 = 0; c2 < 3; ++c2)
          atom_add(&vacc[((long)d * 3 + c2) * Hh + h], tt * ev[c2]);
      }
    }
  }
}

// ---------------------------------------------------------------------------
// LayerNorm over H=128: one 128-thread block per row, LDS tree reduction.
// ---------------------------------------------------------------------------
__global__ void ln_kernel(const float* __restrict__ x, const float* __restrict__ g,
                          const float* __restrict__ b, float* __restrict__ out)
{
  __shared__ float red[Hh];
  const int n = blockIdx.x, t = threadIdx.x;
  const float v = x[(long)n * Hh + t];
  red[t] = v; __syncthreads();
  for (int s = Hh / 2; s > 0; s >>= 1) { if (t < s) red[t] += red[t + s]; __syncthreads(); }
  const float mu = red[0] * (1.0f / Hh);
  __syncthreads();
  const float dv = v - mu;
  red[t] = dv * dv; __syncthreads();
  for (int s = Hh / 2; s > 0; s >>= 1) { if (t < s) red[t] += red[t + s]; __syncthreads(); }
  const float var = red[0] * (1.0f / Hh);
  out[(long)n * Hh + t] = dv * rsqrtf(var + 1e-5f) * g[t] + b[t];
}

__global__ void scale_kernel(float* __restrict__ p, long n)
{
  long i = (long)blockIdx.x * blockDim.x + threadIdx.x;
  if (i < n) p[i] *= INV_SQRT_2f;
}

// vec_dot, vec2_norm, and hcat = [x | vec2_norm]
__global__ void dotnorm_kernel(const float* __restrict__ vp,   // [3N,256]
                               const float* __restrict__ xmsg, // [N,128]
                               float* __restrict__ vdot,       // [N,128]
                               float* __restrict__ hcat)       // [N,256]
{
  long idx = (long)blockIdx.x * blockDim.x + threadIdx.x;
  if (idx >= (long)Nn * Hh) return;
  const int n = (int)(idx >> 7);
  const int h = (int)(idx & 127);
  float dot = 0.0f, nrm = 0.0f;
#pragma unroll
  for (int c = 0; c < 3; ++c) {
    const float* rowp = vp + ((long)(n * 3 + c)) * 256;
    const float a = rowp[h];
    const float b2 = rowp[128 + h];
    dot += a * b2;
    nrm += b2 * b2;
  }
  vdot[idx] = dot * INV_SQRT_Hf;
  hcat[(long)n * 256 + h] = xmsg[idx];
  hcat[(long)n * 256 + 128 + h] = sqrtf(nrm + 1e-8f);
}

// Final epilogue: d_out = [vec_out (N*3*128) | x_out (N*128)]
__global__ void final_kernel(const float* __restrict__ vmsg, const float* __restrict__ xmsg,
                             const float* __restrict__ hout, const float* __restrict__ vp,
                             const float* __restrict__ vdot, float* __restrict__ out)
{
  long idx = (long)blockIdx.x * blockDim.x + threadIdx.x;
  const long nvec = (long)Nn * 384;
  if (idx < nvec) {
    const int n = (int)(idx / 384);
    const int r = (int)(idx % 384);
    const int c = r >> 7, h = r & 127;
    const float xv3 = hout[(long)n * 384 + 256 + h];
    const float v1  = vp[((long)(n * 3 + c)) * 256 + h];
    out[idx] = vmsg[idx] + xv3 * v1;
  } else if (idx < nvec + (long)Nn * Hh) {
    const long j = idx - nvec;
    const int n = (int)(j >> 7), h = (int)(j & 127);
    const float xv1 = hout[(long)n * 384 + h];
    const float xv2 = hout[(long)n * 384 + 128 + h];
    out[idx] = xmsg[j] + (xv1 + xv2 * vdot[j]) * INV_SQRT_2f;
  }
}

extern "C" void kernel_launch(void* const* d_in, const int* in_sizes, int n_in,
                              void* d_out, int out_size, void* d_ws, size_t ws_size,
                              hipStream_t stream)
{
  const float* x    = (const float*)d_in[0];
  const float* vec  = (const float*)d_in[1];
  const int*   eidx = (const int*)  d_in[2];
  const float* eemb = (const float*)d_in[3];
  const float* evec = (const float*)d_in[4];
  const float* ln_g = (const float*)d_in[5];
  const float* ln_b = (const float*)d_in[6];
  const float* W1   = (const float*)d_in[7];
  const float* b1   = (const float*)d_in[8];
  const float* W2   = (const float*)d_in[9];
  const float* b2   = (const float*)d_in[10];
  const float* Wr   = (const float*)d_in[11];
  const float* br   = (const float*)d_in[12];
  const float* Wv   = (const float*)d_in[13];
  const float* Wu1  = (const float*)d_in[14];
  const float* bu1  = (const float*)d_in[15];
  const float* Wu2  = (const float*)d_in[16];
  const float* bu2  = (const float*)d_in[17];

  // Workspace layout (floats); buffers reused across phases.
  float* ws        = (float*)d_ws;
  float* lnx_vdot  = ws;                           // N*128 (lnx, later vec_dot)
  float* mid       = lnx_vdot + (long)Nn * 128;    // N*128 (mlp mid, later hmid)
  float* xh        = mid + (long)Nn * 128;         // N*384 (xh, later hout)
  float* xmsg      = xh + (long)Nn * 384;          // N*128
  float* vmsg      = xmsg + (long)Nn * 128;        // N*384
  float* vp        = vmsg + (long)Nn * 384;        // N*768  ([3N,256])
  float* hcat      = vp + (long)Nn * 768;          // N*256
  (void)ws_size; (void)in_sizes; (void)n_in; (void)out_size;

  // Accumulators start at x / vec (segment_sum adds on top; scale after).
  hipMemcpyAsync(xmsg, x,   (size_t)Nn * 128 * sizeof(float), hipMemcpyDeviceToDevice, stream);
  hipMemcpyAsync(vmsg, vec, (size_t)Nn * 384 * sizeof(float), hipMemcpyDeviceToDevice, stream);

  ln_kernel<<<Nn, Hh, 0, stream>>>(x, ln_g, ln_b, lnx_vdot);

  auto blocks_for = [](long M, long ncol) {
    long tiles = (M / 16) * (ncol / 64);
    return (int)((tiles + 7) / 8);                 // 8 waves per 256-thread block
  };

  // message MLP: xh = silu_s(LN(x)@W1+b1)@W2 + b2
  gemm_wmma_f32<128, 128, 1><<<blocks_for(Nn, 128), 256, 0, stream>>>(lnx_vdot, W1, b1, mid, Nn);
  gemm_wmma_f32<128, 384, 0><<<blocks_for(Nn, 384), 256, 0, stream>>>(mid, W2, b2, xh, Nn);

  // fused edge GEMM + gather/scatter (the dominant phase; L2-resident atomics)
  {
    long tiles = (long)(En / 16) * 6;
    int blocks = (int)((tiles + 7) / 8);
    edge_msg_wmma<<<blocks, 256, 0, stream>>>(eemb, Wr, br, eidx, evec, xh, vec, xmsg, vmsg);
  }

  scale_kernel<<<(int)(((long)Nn * 128 + 255) / 256), 256, 0, stream>>>(xmsg, (long)Nn * 128);

  // update: vproj = vec_new @ Wv  (treat vec as [3N,128] row-major)
  gemm_wmma_f32<128, 256, 0><<<blocks_for((long)Nn * 3, 256), 256, 0, stream>>>(vmsg, Wv, nullptr, vp, Nn * 3);

  dotnorm_kernel<<<(int)(((long)Nn * 128 + 255) / 256), 256, 0, stream>>>(vp, xmsg, lnx_vdot, hcat);

  // update MLP: hout = silu_s(hcat@Wu1+bu1)@Wu2 + bu2
  gemm_wmma_f32<256, 128, 1><<<blocks_for(Nn, 128), 256, 0, stream>>>(hcat, Wu1, bu1, mid, Nn);
  gemm_wmma_f32<128, 384, 0><<<blocks_for(Nn, 384), 256, 0, stream>>>(mid, Wu2, bu2, xh, Nn);

  {
    long total = (long)Nn * 512;                  // N*384 vec_out + N*128 x_out
    final_kernel<<<(int)((total + 255) / 256), 256, 0, stream>>>(
        vmsg, xmsg, xh, vp, lnx_vdot, (float*)d_out);
  }
}